// BetaTCVAE_37752762532556
// MI455X (gfx1250) — compile-verified
//
#include <hip/hip_runtime.h>
#include <math.h>

// BetaTCVAE loss, MI455X (gfx1250, wave32).
// BATCH=2048, LATENT=64. Joint path via fp32 WMMA GEMM (K=128 feature trick),
// per-dim path via online logsumexp on VALU/trans units. All fp32.

#define BATCH   2048
#define LATENT  64
#define KDIM    128           // 2 features per latent dim
#define LOG2PI  1.83787706640934533908193770912476f
#define BETA    6.0f

typedef __attribute__((ext_vector_type(2))) float v2f;
typedef __attribute__((ext_vector_type(8))) float v8f;

__device__ __forceinline__ float wave_sum32(float v) {
#pragma unroll
  for (int off = 16; off > 0; off >>= 1) v += __shfl_xor(v, off, 32);
  return v;
}

__device__ __forceinline__ void lse_update(float& m, float& s, float a) {
  float mn = fmaxf(m, a);
  s = s * __expf(m - mn) + __expf(a - mn);
  m = mn;
}

// ---------------------------------------------------------------------------
// Kernel 1: per-row prep. One wave per row (row used as both i and j index).
// Lane handles l = lane and l = lane+32.
//   Af[i, 2l]   = z^2          Af[i, 2l+1] = z
//   Bf[j, 2l]   = w=exp(-lv)   Bf[j, 2l+1] = -2*m*w
//   D[j]  = sum_l (m^2 w + lv) + 64*log(2pi)      (per-j logit constant)
//   KL[j] = sum_l (m^2 + exp(lv) - lv - 1)
//   Wt/Ut/Vt: [64, 2048] transposed arrays for the per-dim path:
//   Wt=w, Ut=m*w, Vt=m^2 w + lv + log(2pi)
// ---------------------------------------------------------------------------
__global__ __launch_bounds__(256) void prep_kernel(
    const float* __restrict__ z, const float* __restrict__ zm,
    const float* __restrict__ zlv,
    float* __restrict__ Af, float* __restrict__ Bf,
    float* __restrict__ D,  float* __restrict__ KL,
    float* __restrict__ Wt, float* __restrict__ Ut, float* __restrict__ Vt) {
  int gtid = blockIdx.x * 256 + threadIdx.x;
  int row  = gtid >> 5;          // 0..2047
  int lane = gtid & 31;
  float dsum = 0.f, klsum = 0.f;
#pragma unroll
  for (int t = 0; t < 2; ++t) {
    int l = lane + 32 * t;
    int idx = row * LATENT + l;
    float zv = z[idx];
    float mv = zm[idx];
    float lv = zlv[idx];
    float w  = __expf(-lv);
    float mw = mv * w;
    Af[row * KDIM + 2 * l]     = zv * zv;
    Af[row * KDIM + 2 * l + 1] = zv;
    Bf[row * KDIM + 2 * l]     = w;
    Bf[row * KDIM + 2 * l + 1] = -2.0f * mw;
    Wt[l * BATCH + row] = w;
    Ut[l * BATCH + row] = mw;
    Vt[l * BATCH + row] = fmaf(mv, mw, lv + LOG2PI);
    dsum  += fmaf(mv, mw, lv);                       // m^2 w + lv
    klsum += mv * mv + __expf(lv) - lv - 1.0f;
  }
  dsum  = wave_sum32(dsum);
  klsum = wave_sum32(klsum);
  if (lane == 0) {
    D[row]  = dsum + (float)LATENT * LOG2PI;
    KL[row] = klsum;
  }
}

// ---------------------------------------------------------------------------
// Kernel 2: WMMA fp32 GEMM. One wave computes one 16x16 tile of
//   S[i,j] = sum_k Af[i,k] * Bf[j,k]     (K = 128)
// using V_WMMA_F32_16X16X4_F32 (32 per tile), then fused epilogue stores the
// final logits T[i,j] = -0.5 * (S[i,j] + D[j]).
// ISA layout (16x4 fp32 A, wave32): lanes 0-15 hold M=lane with K=0 (v0)/K=1
// (v1); lanes 16-31 hold M=lane-16 with K=2/K=3. B mirrors with N in place of
// M. Both operands therefore load a float2 at (row, k0 + 2*(lane>=16)).
// ---------------------------------------------------------------------------
__global__ __launch_bounds__(256) void tile_gemm(
    const float* __restrict__ Af, const float* __restrict__ Bf,
    const float* __restrict__ D, float* __restrict__ T) {
  int gtid = blockIdx.x * 256 + threadIdx.x;
  int wave = gtid >> 5;          // 0..16383
  int lane = gtid & 31;
  int ti = wave >> 7;            // tile row   (i)
  int tj = wave & 127;           // tile col   (j)
  int i0 = ti << 4, j0 = tj << 4;
  int r  = lane & 15;
  int kh = (lane >> 4) << 1;     // 0 or 2
  const float* ap = Af + (i0 + r) * KDIM + kh;
  const float* bp = Bf + (j0 + r) * KDIM + kh;
  v8f c = {};
#pragma unroll
  for (int k0 = 0; k0 < KDIM; k0 += 4) {
    v2f a = *(const v2f*)(ap + k0);
    v2f b = *(const v2f*)(bp + k0);
    // (neg_a, A, neg_b, B, c_mod, C, reuse_a, reuse_b)
    c = __builtin_amdgcn_wmma_f32_16x16x4_f32(false, a, false, b,
                                              (short)0, c, false, false);
  }
  // C/D layout: VGPR v, lanes 0-15 -> M=v, N=lane; lanes 16-31 -> M=v+8.
  float dj    = D[j0 + r];
  int   mbase = i0 + ((lane >> 4) << 3);
#pragma unroll
  for (int v = 0; v < 8; ++v) {
    T[(mbase + v) * BATCH + (j0 + r)] = -0.5f * (c[v] + dj);
  }
}

// ---------------------------------------------------------------------------
// Kernel 3: joint logsumexp. One wave per i, streaming logsumexp over the
// 2048-entry row of T, lanes stride by 32, then shuffle-combine.
// ---------------------------------------------------------------------------
__global__ __launch_bounds__(256) void lse_rows(
    const float* __restrict__ T, float* __restrict__ lj) {
  int gtid = blockIdx.x * 256 + threadIdx.x;
  int i    = gtid >> 5;          // 0..2047
  int lane = gtid & 31;
  const float* row = T + i * BATCH;
  float m = -INFINITY, s = 0.f;
  for (int j = lane; j < BATCH; j += 32) lse_update(m, s, row[j]);
#pragma unroll
  for (int off = 16; off > 0; off >>= 1) {
    float mo = __shfl_xor(m, off, 32);
    float so = __shfl_xor(s, off, 32);
    float mn = fmaxf(m, mo);
    s = s * __expf(m - mn) + so * __expf(mo - mn);
    m = mn;
  }
  if (lane == 0) lj[i] = m + __logf(s);
}

// ---------------------------------------------------------------------------
// Kernel 4: per-dim logsumexp (the 268M-exp hot loop). One wave per (i,l):
//   a_ij = -0.5 * (z^2 * Wt[l,j] - 2 z * Ut[l,j] + Vt[l,j])
// lanes read consecutive j (coalesced; Wt/Ut/Vt are 1.5MB -> L2 resident).
// ---------------------------------------------------------------------------
__global__ __launch_bounds__(256) void lse_dims(
    const float* __restrict__ z,
    const float* __restrict__ Wt, const float* __restrict__ Ut,
    const float* __restrict__ Vt, float* __restrict__ lse2) {
  int gtid = blockIdx.x * 256 + threadIdx.x;
  int wave = gtid >> 5;          // 0..131071
  int lane = gtid & 31;
  int i = wave >> 6;
  int l = wave & 63;
  float zv = z[i * LATENT + l];
  float z2 = zv * zv;
  float nz = -2.0f * zv;
  const float* wr = Wt + l * BATCH;
  const float* ur = Ut + l * BATCH;
  const float* vr = Vt + l * BATCH;
  float m = -INFINITY, s = 0.f;
  for (int j = lane; j < BATCH; j += 32) {
    float a = -0.5f * fmaf(z2, wr[j], fmaf(nz, ur[j], vr[j]));
    lse_update(m, s, a);
  }
#pragma unroll
  for (int off = 16; off > 0; off >>= 1) {
    float mo = __shfl_xor(m, off, 32);
    float so = __shfl_xor(s, off, 32);
    float mn = fmaxf(m, mo);
    s = s * __expf(m - mn) + so * __expf(mo - mn);
    m = mn;
  }
  if (lane == 0) lse2[i * LATENT + l] = m + __logf(s);
}

// ---------------------------------------------------------------------------
// Kernel 5: final scalar. Deterministic fixed-tree reduction.
//   out = (BETA-1) * mean_i(lj[i] - sum_l lse2[i,l]) + 0.5 * mean_i KL[i]
// ---------------------------------------------------------------------------
__global__ __launch_bounds__(256) void final_reduce(
    const float* __restrict__ lj, const float* __restrict__ lse2,
    const float* __restrict__ KL, float* __restrict__ out) {
  __shared__ float s1[256];
  __shared__ float s2[256];
  float tc = 0.f, kl = 0.f;
  for (int i = threadIdx.x; i < BATCH; i += 256) {
    float p = 0.f;
#pragma unroll 8
    for (int l = 0; l < LATENT; ++l) p += lse2[i * LATENT + l];
    tc += lj[i] - p;
    kl += KL[i];
  }
  s1[threadIdx.x] = tc;
  s2[threadIdx.x] = kl;
  __syncthreads();
#pragma unroll
  for (int off = 128; off > 0; off >>= 1) {
    if (threadIdx.x < off) {
      s1[threadIdx.x] += s1[threadIdx.x + off];
      s2[threadIdx.x] += s2[threadIdx.x + off];
    }
    __syncthreads();
  }
  if (threadIdx.x == 0) {
    out[0] = (BETA - 1.0f) * (s1[0] / (float)BATCH)
           + 0.5f * (s2[0] / (float)BATCH);
  }
}

// ---------------------------------------------------------------------------
extern "C" void kernel_launch(void* const* d_in, const int* in_sizes, int n_in,
                              void* d_out, int out_size, void* d_ws,
                              size_t ws_size, hipStream_t stream) {
  (void)in_sizes; (void)n_in; (void)out_size; (void)ws_size;
  const float* z   = (const float*)d_in[0];
  const float* zm  = (const float*)d_in[1];
  const float* zlv = (const float*)d_in[2];

  // Workspace layout (floats); total ~5.25M floats (~21 MB).
  float* ws = (float*)d_ws;
  float* Af = ws;                          // 2048*128
  float* Bf = Af + BATCH * KDIM;           // 2048*128
  float* D  = Bf + BATCH * KDIM;           // 2048
  float* KL = D  + BATCH;                  // 2048
  float* Wt = KL + BATCH;                  // 64*2048
  float* Ut = Wt + LATENT * BATCH;         // 64*2048
  float* Vt = Ut + LATENT * BATCH;         // 64*2048
  float* T  = Vt + LATENT * BATCH;         // 2048*2048
  float* LJ = T  + BATCH * BATCH;          // 2048
  float* L2 = LJ + BATCH;                  // 2048*64

  prep_kernel <<<  256, 256, 0, stream>>>(z, zm, zlv, Af, Bf, D, KL, Wt, Ut, Vt);
  tile_gemm   <<< 2048, 256, 0, stream>>>(Af, Bf, D, T);
  lse_rows    <<<  256, 256, 0, stream>>>(T, LJ);
  lse_dims    <<<16384, 256, 0, stream>>>(z, Wt, Ut, Vt, L2);
  final_reduce<<<    1, 256, 0, stream>>>(LJ, L2, KL, (float*)d_out);
}